// SFC_CAE_Decoder_1460288880656
// MI455X (gfx1250) — compile-verified
//
#include <hip/hip_runtime.h>
#include <math.h>

// CDNA5 / gfx1250, wave32.
// Pipeline:
//  K1: h1 = tanh(x @ W1^T + b1)              [16,512]    (WMMA f32 16x16x4)
//  K2: h2 = tanh(h1 @ W2^T + b2)             [16,131104] (WMMA f32 16x16x4, HBM-bound on W2)
//  K3: per-sfc ConvT1d 16->16 s4 K32 p16 op1 + tanh  -> c0 [2,16,16,16385]
//      (input window staged to LDS via global_load_async_to_lds_b32)
//  K4: per-sfc ConvT1d 16->1  s4 K32 p16 op0 + tanh  -> c1 [2,16,65536]
//  K5: SFC 3-neighbour gather + tanh, cross-SFC combine + tanh, output affine

typedef __attribute__((ext_vector_type(2))) float v2f;
typedef __attribute__((ext_vector_type(8))) float v8f;

// ---- async global->LDS staging (CDNA5 ASYNCcnt path), with safe fallback ----
#if defined(__has_builtin)
#if __has_builtin(__builtin_amdgcn_global_load_async_to_lds_b32) && \
    __has_builtin(__builtin_amdgcn_s_wait_asynccnt)
#define USE_ASYNC_LDS 1
#endif
#endif

#if defined(USE_ASYNC_LDS)
// b32 flavor takes typed int pointers: (AS1 int* src, AS3 int* dst, imm, imm)
#define ASYNC_STAGE_B32(gsrc, ldst)                                         \
  __builtin_amdgcn_global_load_async_to_lds_b32(                            \
      (__attribute__((address_space(1))) int*)(gsrc),                       \
      (__attribute__((address_space(3))) int*)(ldst), 0, 0)
#define ASYNC_WAIT() __builtin_amdgcn_s_wait_asynccnt(0)
#else
#define ASYNC_STAGE_B32(gsrc, ldst) (*(ldst) = *(gsrc))
#define ASYNC_WAIT()
#endif

// out[m][n] = tanh( sum_k A[m][k]*Bm[n][k] + bias[n] ),  m in [0,16)
// One wave computes one 16-wide N tile via V_WMMA_F32_16X16X4_F32.
// A-matrix VGPR layout (16x4 f32): lanes 0-15 / 16-31 hold K-halves {0,1}/{2,3};
// B-matrix (4x16) mirrors it with lanes indexing N. Since Bm is row-major [n][k],
// the transposed-B read pattern is identical to the A read pattern.
__global__ __launch_bounds__(256) void k_gemm_tanh(
    const float* __restrict__ A, const float* __restrict__ Bm,
    const float* __restrict__ bias, float* __restrict__ out,
    int K, int N, int ntiles)
{
  int wave = blockIdx.x * (blockDim.x >> 5) + (threadIdx.x >> 5);
  if (wave >= ntiles) return;                 // wave-uniform: EXEC stays all-1s
  int lane = threadIdx.x & 31;
  int m    = lane & 15;
  int half = lane >> 4;                       // 0: K={0,1}, 1: K={2,3}
  int koff = half << 1;
  int n    = (wave << 4) + m;                 // column this lane owns in B/C/D

  const float* arow = A  + (size_t)m * K;
  const float* brow = Bm + (size_t)n * K;
  __builtin_prefetch(brow, 0, 1);             // global_prefetch_b8: warm the W row
  __builtin_prefetch(brow + (K >> 1), 0, 1);

  v8f acc = {};
  for (int kk = 0; kk < K; kk += 4) {
    v2f a = *(const v2f*)(arow + kk + koff);  // 8B aligned (kk,koff even)
    v2f b = *(const v2f*)(brow + kk + koff);
    acc = __builtin_amdgcn_wmma_f32_16x16x4_f32(
        /*neg_a=*/false, a, /*neg_b=*/false, b,
        /*c_mod=*/(short)0, acc, /*reuse_a=*/false, /*reuse_b=*/false);
  }

  float bv = bias[n];
  int mbase = half << 3;                      // D: VGPR v -> row v (lanes 0-15) / v+8
  #pragma unroll
  for (int v = 0; v < 8; ++v) {
    out[(size_t)(mbase + v) * N + n] = tanhf(acc[v] + bv);
  }
}

// ConvTranspose1d: Cin=16, Cout=16, Lin=4097, Lout=16385, stride=4, K=32, pad=16.
// Input channel c, pos l of sfc i lives at h2[b*131104 + (c*4097+l)*2 + i].
// Block = (sfc, batch, cout, 256-wide o tile). Shared input window 72 x 16ci
// is staged to LDS with async loads; OOB taps pre-zeroed.
__global__ __launch_bounds__(256) void k_convt0_tanh(
    const float* __restrict__ h2, const float* __restrict__ ctw0,
    const float* __restrict__ ctb0, float* __restrict__ c0)
{
  __shared__ float ws[512];                   // w[ci][k] for this (sfc, co)
  __shared__ float sin_s[16 * 72];            // input window [ci][l - lbase]
  int bid = blockIdx.x;
  int ob = bid % 65;
  int co = (bid / 65) & 15;
  int bb = (bid / (65 * 16)) & 15;
  int i  = bid / (65 * 16 * 16);
  int tid = threadIdx.x;
  int o0 = ob * 256;
  int lbase = (o0 >> 2) - 3;                  // lowest l any o in tile can touch

  for (int t = tid; t < 512; t += 256) {
    int ci = t >> 5, k = t & 31;
    ws[t] = ctw0[(((i * 16 + ci) * 16) + co) * 32 + k];
  }
  const float* in = h2 + bb * 131104 + i;
  for (int t = tid; t < 16 * 72; t += 256) {
    int ci = t / 72, dl = t - ci * 72;
    int l = lbase + dl;
    if (l >= 0 && l <= 4096) {
      ASYNC_STAGE_B32(in + (ci * 4097 + l) * 2, &sin_s[t]);
    } else {
      sin_s[t] = 0.f;
    }
  }
  ASYNC_WAIT();
  __syncthreads();

  int o = o0 + tid;
  if (o >= 16385) return;
  float acc = ctb0[i * 16 + co];
  int lmax = (o + 16) >> 2;
  int p    = (o + 16) & 3;
  int dl0  = lmax - 7 - lbase;                // in [0,63]
  #pragma unroll
  for (int j = 0; j < 8; ++j) {
    int dl = dl0 + j;
    int k  = p + ((7 - j) << 2);
    #pragma unroll
    for (int ci = 0; ci < 16; ++ci)
      acc += sin_s[ci * 72 + dl] * ws[ci * 32 + k];
  }
  c0[(((i * 16 + bb) * 16) + co) * 16385 + o] = tanhf(acc);
}

// ConvTranspose1d: Cin=16, Cout=1, Lin=16385, Lout=65536, stride=4, K=32, pad=16.
__global__ __launch_bounds__(256) void k_convt1_tanh(
    const float* __restrict__ c0, const float* __restrict__ ctw1,
    const float* __restrict__ ctb1, float* __restrict__ c1)
{
  __shared__ float ws[512];
  __shared__ float sin_s[16 * 72];
  int bid = blockIdx.x;
  int ob = bid & 255;
  int bb = (bid >> 8) & 15;
  int i  = bid >> 12;
  int tid = threadIdx.x;
  int o0 = ob * 256;
  int lbase = (o0 >> 2) - 3;

  for (int t = tid; t < 512; t += 256)
    ws[t] = ctw1[(i * 16 + (t >> 5)) * 32 + (t & 31)];
  const float* in = c0 + ((i * 16 + bb) * 16) * 16385;
  for (int t = tid; t < 16 * 72; t += 256) {
    int ci = t / 72, dl = t - ci * 72;
    int l = lbase + dl;
    if (l >= 0 && l <= 16384) {
      ASYNC_STAGE_B32(in + ci * 16385 + l, &sin_s[t]);
    } else {
      sin_s[t] = 0.f;
    }
  }
  ASYNC_WAIT();
  __syncthreads();

  int o = o0 + tid;
  float acc = ctb1[i];
  int lmax = (o + 16) >> 2;
  int p    = (o + 16) & 3;
  int dl0  = lmax - 7 - lbase;
  #pragma unroll
  for (int j = 0; j < 8; ++j) {
    int dl = dl0 + j;
    int k  = p + ((7 - j) << 2);
    #pragma unroll
    for (int ci = 0; ci < 16; ++ci)
      acc += sin_s[ci * 72 + dl] * ws[ci * 32 + k];
  }
  c1[(i * 16 + bb) * 65536 + o] = tanhf(acc);
}

// Fused: per-sfc (minus,self,plus) gather + weighted sum + tanh, then
// cross-SFC weighted sum + tanh, then out = z*out_w + out_b.
__global__ __launch_bounds__(256) void k_sps_final(
    const float* __restrict__ c1,
    const float* __restrict__ sps_w, const float* __restrict__ sps_b,
    const float* __restrict__ final_w, const float* __restrict__ final_b,
    const float* __restrict__ out_w, const float* __restrict__ out_b,
    const int* __restrict__ ord_idx, const int* __restrict__ plus_idx,
    const int* __restrict__ minus_idx, float* __restrict__ out)
{
  int gid = blockIdx.x * 256 + threadIdx.x;
  int n  = gid & 65535;
  int bb = gid >> 16;
  float zacc = final_b[n];
  #pragma unroll
  for (int i = 0; i < 2; ++i) {
    int idx = i * 65536 + n;
    const float* bbase = c1 + (i * 16 + bb) * 65536;
    float v = bbase[minus_idx[idx]] * sps_w[idx * 3 + 0]
            + bbase[ord_idx[idx]]   * sps_w[idx * 3 + 1]
            + bbase[plus_idx[idx]]  * sps_w[idx * 3 + 2]
            + sps_b[idx];
    zacc += tanhf(v) * final_w[n * 2 + i];
  }
  float z = tanhf(zacc);
  out[bb * 65536 + n] = z * out_w[n] + out_b[n];
}

extern "C" void kernel_launch(void* const* d_in, const int* in_sizes, int n_in,
                              void* d_out, int out_size, void* d_ws, size_t ws_size,
                              hipStream_t stream)
{
  (void)in_sizes; (void)n_in; (void)out_size; (void)ws_size;
  const float* x    = (const float*)d_in[0];
  const float* W1   = (const float*)d_in[1];
  const float* b1   = (const float*)d_in[2];
  const float* W2   = (const float*)d_in[3];
  const float* b2   = (const float*)d_in[4];
  const float* ctw0 = (const float*)d_in[5];
  const float* ctb0 = (const float*)d_in[6];
  const float* ctw1 = (const float*)d_in[7];
  const float* ctb1 = (const float*)d_in[8];
  const float* spsw = (const float*)d_in[9];
  const float* spsb = (const float*)d_in[10];
  const float* fw   = (const float*)d_in[11];
  const float* fb   = (const float*)d_in[12];
  const float* ow   = (const float*)d_in[13];
  const float* ob   = (const float*)d_in[14];
  const int* ordi   = (const int*)d_in[15];
  const int* plusi  = (const int*)d_in[16];
  const int* minusi = (const int*)d_in[17];
  float* out = (float*)d_out;

  // Workspace layout (floats): total 12,592,128 floats = 50.4 MB
  float* h1 = (float*)d_ws;        // 16*512          =     8,192
  float* h2 = h1 + 8192;           // 16*131104       = 2,097,664
  float* c0 = h2 + 2097664;        // 2*16*16*16385   = 8,389,120
  float* c1 = c0 + 8389120;        // 2*16*65536      = 2,097,152

  // K1: [16,128]x[128,512] -> h1.  32 N-tiles, 8 waves/block -> 4 blocks.
  k_gemm_tanh<<<4, 256, 0, stream>>>(x, W1, b1, h1, 128, 512, 32);
  // K2: [16,512]x[512,131104] -> h2.  8194 N-tiles -> 1025 blocks.
  k_gemm_tanh<<<1025, 256, 0, stream>>>(h1, W2, b2, h2, 512, 131104, 8194);
  // K3: 2 sfc x 16 batch x 16 cout x ceil(16385/256)=65 o-blocks
  k_convt0_tanh<<<2 * 16 * 16 * 65, 256, 0, stream>>>(h2, ctw0, ctb0, c0);
  // K4: 2 sfc x 16 batch x 256 o-blocks
  k_convt1_tanh<<<2 * 16 * 256, 256, 0, stream>>>(c0, ctw1, ctb1, c1);
  // K5: 16*65536 elements / 256
  k_sps_final<<<4096, 256, 0, stream>>>(c1, spsw, spsb, fw, fb, ow, ob,
                                        ordi, plusi, minusi, out);
}